// DeepAggregateAutoEncoder_11149735100496
// MI455X (gfx1250) — compile-verified
//
#include <hip/hip_runtime.h>

// Problem geometry (fixed by the reference).
#define NB    1024            // batch rows
#define F0    4096            // input width
#define L0N   2048
#define L1N   1024
#define L2N   2048
#define L3N   4096
#define CC    64              // gather fan-in
#define ACTW  (L0N + L1N + L2N)   // 5120
#define NT    256             // threads / block (8 wave32 waves)
#define ROWS  2               // batch rows per block (index reuse)

__global__ __launch_bounds__(NT) void daa_all_layers(
    const float* __restrict__ x,
    const int* __restrict__ conn0, const int* __restrict__ conn1,
    const int* __restrict__ conn2, const int* __restrict__ conn3,
    const int* __restrict__ op0, const int* __restrict__ op1,
    const int* __restrict__ op2, const int* __restrict__ op3,
    float* __restrict__ out_x, float* __restrict__ out_act)
{
    __shared__ __align__(16) float bufA[ROWS][F0];   // 32 KB
    __shared__ __align__(16) float bufB[ROWS][F0];   // 32 KB

    const int t    = threadIdx.x;
    const int row0 = blockIdx.x * ROWS;

    // ---- Stage ROWS input rows (2 x 16 KB) into LDS via CDNA5 async DMA ----
    {
        const unsigned ldsA = (unsigned)(size_t)&bufA[0][0];
        #pragma unroll
        for (int r = 0; r < ROWS; ++r) {
            const unsigned long long g =
                (unsigned long long)(const void*)(x + (size_t)(row0 + r) * F0);
            #pragma unroll
            for (int k = 0; k < (F0 * 4) / (NT * 16); ++k) {   // 4 chunks/row
                const unsigned off = (unsigned)((k * NT + t) * 16);
                const unsigned lds_addr = ldsA + (unsigned)(r * F0 * 4) + off;
                const unsigned long long ga = g + off;
                // 16 bytes per lane, global -> LDS, tracked on ASYNCcnt
                asm volatile("global_load_async_to_lds_b128 %0, %1, off"
                             :: "v"(lds_addr), "v"(ga) : "memory");
            }
        }
#if __has_builtin(__builtin_amdgcn_s_wait_asynccnt)
        __builtin_amdgcn_s_wait_asynccnt(0);
#else
        asm volatile("s_wait_asynccnt 0" ::: "memory");
#endif
    }
    __syncthreads();

    float* act0 = out_act + (size_t)(row0 + 0) * ACTW;
    float* act1 = out_act + (size_t)(row0 + 1) * ACTW;

    // One aggregate layer: for each output neuron j, gather CC values from the
    // LDS-resident input row(s) and reduce with min/max (op-selected).
    // Index quads are loaded once and reused for both batch rows.
#define DAA_LAYER(IN, CONN, OPS, OUTN, STORE)                                   \
    for (int j = t; j < (OUTN); j += NT) {                                      \
        const int4* cp = (const int4*)((CONN) + (size_t)j * CC);                \
        float mn0 = 3.402823466e+38f, mx0 = -3.402823466e+38f;                  \
        float mn1 = 3.402823466e+38f, mx1 = -3.402823466e+38f;                  \
        _Pragma("unroll")                                                       \
        for (int q = 0; q < CC / 4; ++q) {                                      \
            const int4 id = cp[q];                                              \
            const float a0 = IN[0][id.x], a1 = IN[0][id.y];                     \
            const float a2 = IN[0][id.z], a3 = IN[0][id.w];                     \
            const float c0v = IN[1][id.x], c1v = IN[1][id.y];                   \
            const float c2v = IN[1][id.z], c3v = IN[1][id.w];                   \
            mn0 = fminf(mn0, fminf(fminf(a0, a1), fminf(a2, a3)));              \
            mx0 = fmaxf(mx0, fmaxf(fmaxf(a0, a1), fmaxf(a2, a3)));              \
            mn1 = fminf(mn1, fminf(fminf(c0v, c1v), fminf(c2v, c3v)));          \
            mx1 = fmaxf(mx1, fmaxf(fmaxf(c0v, c1v), fmaxf(c2v, c3v)));          \
        }                                                                       \
        const int o = (OPS)[j];                                                 \
        const float r0 = (o == 0) ? mn0 : mx0;                                  \
        const float r1 = (o == 0) ? mn1 : mx1;                                  \
        STORE;                                                                  \
    }

    // L0: bufA -> bufB ; hidden acts @ [0, 2048)
    DAA_LAYER(bufA, conn0, op0, L0N, {
        bufB[0][j] = r0; bufB[1][j] = r1;
        __builtin_nontemporal_store(r0, &act0[j]);
        __builtin_nontemporal_store(r1, &act1[j]);
    })
    __syncthreads();

    // L1: bufB -> bufA ; hidden acts @ [2048, 3072)
    DAA_LAYER(bufB, conn1, op1, L1N, {
        bufA[0][j] = r0; bufA[1][j] = r1;
        __builtin_nontemporal_store(r0, &act0[L0N + j]);
        __builtin_nontemporal_store(r1, &act1[L0N + j]);
    })
    __syncthreads();

    // L2: bufA -> bufB ; hidden acts @ [3072, 5120)
    DAA_LAYER(bufA, conn2, op2, L2N, {
        bufB[0][j] = r0; bufB[1][j] = r1;
        __builtin_nontemporal_store(r0, &act0[L0N + L1N + j]);
        __builtin_nontemporal_store(r1, &act1[L0N + L1N + j]);
    })
    __syncthreads();

    // L3: bufB -> out_x (reconstruction), global only
    float* xo0 = out_x + (size_t)(row0 + 0) * L3N;
    float* xo1 = out_x + (size_t)(row0 + 1) * L3N;
    DAA_LAYER(bufB, conn3, op3, L3N, {
        __builtin_nontemporal_store(r0, &xo0[j]);
        __builtin_nontemporal_store(r1, &xo1[j]);
    })
#undef DAA_LAYER
}

extern "C" void kernel_launch(void* const* d_in, const int* in_sizes, int n_in,
                              void* d_out, int out_size, void* d_ws, size_t ws_size,
                              hipStream_t stream) {
    (void)in_sizes; (void)n_in; (void)d_ws; (void)ws_size; (void)out_size;
    const float* x  = (const float*)d_in[0];
    const int*   c0 = (const int*)d_in[1];
    const int*   c1 = (const int*)d_in[2];
    const int*   c2 = (const int*)d_in[3];
    const int*   c3 = (const int*)d_in[4];
    const int*   o0 = (const int*)d_in[5];
    const int*   o1 = (const int*)d_in[6];
    const int*   o2 = (const int*)d_in[7];
    const int*   o3 = (const int*)d_in[8];

    float* out_x   = (float*)d_out;                        // (1024, 4096)
    float* out_act = out_x + (size_t)NB * L3N;             // (1024, 5120)

    daa_all_layers<<<dim3(NB / ROWS), dim3(NT), 0, stream>>>(
        x, c0, c1, c2, c3, o0, o1, o2, o3, out_x, out_act);
}